// Abstract_RC_83391085019192
// MI455X (gfx1250) — compile-verified
//
#include <hip/hip_runtime.h>
#include <cmath>

// Problem constants (from reference)
#define BDIM 16
#define NDIM 1024
#define TDIM 4096
#define ODIM 256
#define SLOPE 2.0f

// Launch geometry: persistent kernel, must be fully resident for grid barrier.
#define NWG_REC 64                 // 64 tiles of 16 columns of h  (N=1024)
#define NWG_PROJ 16                // 16 tiles of 16 columns of out (OUT=256)
#define NWG (NWG_REC + NWG_PROJ)   // 80 workgroups
#define WGSIZE 128                 // 4 wave32s per WG
#define KCHUNK (NDIM / 4)          // each wave covers 256 of K=1024

typedef float v2f __attribute__((ext_vector_type(2)));
typedef float v8f __attribute__((ext_vector_type(8)));

// --- fp32 WMMA: D = A(16x4) * B(4x16) + C(16x16) -------------------------
__device__ __forceinline__ v8f wmma_f32(v2f a, v2f b, v8f c) {
  // 8-arg pattern: (neg_a, A, neg_b, B, c_mod, C, reuse_a, reuse_b)
  return __builtin_amdgcn_wmma_f32_16x16x4_f32(false, a, false, b,
                                               (short)0, c, false, false);
}

// --- workspace init: zero h double-buffer + barrier counter --------------
__global__ void rc_init(float* hbuf, unsigned* cnt) {
  int i = blockIdx.x * blockDim.x + threadIdx.x;
  if (i < 2 * BDIM * NDIM) hbuf[i] = 0.0f;
  if (i == 0) *cnt = 0u;
}

// --- device-wide barrier (all NWG blocks resident) -----------------------
__device__ __forceinline__ void grid_barrier(volatile unsigned* cnt,
                                             unsigned target) {
  __syncthreads();
  if (threadIdx.x == 0) {
    __threadfence();                       // release our global stores
    atomicAdd((unsigned*)cnt, 1u);
    while (*cnt < target) __builtin_amdgcn_s_sleep(1);
  }
  __syncthreads();
  __threadfence();                         // acquire other WGs' stores
}

// --- projection of one 16-wide column tile of out[:, :, tout] ------------
__device__ __forceinline__ void proj_tile(
    const float* __restrict__ hsrc, const float* __restrict__ lw, float lbv,
    float* __restrict__ out, int c0, int k0, int wave, int lane, int lo,
    int hi, int tout, float red[4][8][32]) {
  v8f acc = {0.f, 0.f, 0.f, 0.f, 0.f, 0.f, 0.f, 0.f};
  // A = h [16(b) x 1024], B = lin_w^T [1024 x 256]; both frags are float2 loads.
  const float* arow = hsrc + lo * NDIM + 2 * hi;
  const float* brow = lw + (c0 + lo) * NDIM + 2 * hi;
#pragma unroll 4
  for (int k = 0; k < KCHUNK; k += 4) {
    v2f a = *(const v2f*)(arow + k0 + k);
    v2f b = *(const v2f*)(brow + k0 + k);
    acc = wmma_f32(a, b, acc);
  }
#pragma unroll
  for (int r = 0; r < 8; ++r) red[wave][r][lane] = acc[r];
  __syncthreads();
#pragma unroll
  for (int rr = 0; rr < 2; ++rr) {
    const int r = wave * 2 + rr;
    float v = red[0][r][lane] + red[1][r][lane] + red[2][r][lane] +
              red[3][r][lane];
    const int m = r + 8 * hi;       // batch row
    const int o = c0 + lo;          // output feature
    out[(m * ODIM + o) * TDIM + tout] = v + lbv;
  }
}

// --- fused persistent kernel ---------------------------------------------
__global__ __launch_bounds__(WGSIZE) void rc_fused(
    const float* __restrict__ x, const float* __restrict__ W,
    const float* __restrict__ lw, const float* __restrict__ lb,
    float* __restrict__ out, float* __restrict__ hbuf,
    volatile unsigned* cnt) {
  const int wg = blockIdx.x;
  const int tid = threadIdx.x;
  const int wave = tid >> 5;
  const int lane = tid & 31;
  const int lo = lane & 15;
  const int hi = lane >> 4;
  const bool isRec = (wg < NWG_REC);
  const int tile = isRec ? wg : (wg - NWG_REC);
  const int c0 = tile * 16;        // column base: n (recurrence) or o (proj)
  const int k0 = wave * KCHUNK;    // this wave's K slice

  __shared__ float red[4][8][32];  // cross-wave partial-C reduction

  float lbv = 0.0f;
  if (!isRec) lbv = lb[c0 + lo];

  for (int t = 0; t < TDIM; ++t) {
    const int cur = t & 1;
    const int nxt = cur ^ 1;
    const float* hcur = hbuf + cur * (BDIM * NDIM);  // holds res_{t-1}

    if (isRec) {
      // h_next tile: A = h (16x1024), B = W_res (1024x1024) row-major.
      v8f acc = {0.f, 0.f, 0.f, 0.f, 0.f, 0.f, 0.f, 0.f};
      const float* arow = hcur + lo * NDIM + 2 * hi;
      const float* bcol = W + (2 * hi) * NDIM + c0 + lo;
#pragma unroll 4
      for (int k = 0; k < KCHUNK; k += 4) {
        v2f a = *(const v2f*)(arow + k0 + k);
        v2f b;
        b.x = bcol[(k0 + k) * NDIM];
        b.y = bcol[(k0 + k) * NDIM + NDIM];
        acc = wmma_f32(a, b, acc);
      }
#pragma unroll
      for (int r = 0; r < 8; ++r) red[wave][r][lane] = acc[r];
      __syncthreads();
#pragma unroll
      for (int rr = 0; rr < 2; ++rr) {
        const int r = wave * 2 + rr;
        float v = red[0][r][lane] + red[1][r][lane] + red[2][r][lane] +
                  red[3][r][lane];
        const int m = r + 8 * hi;   // batch row
        const int n = c0 + lo;      // reservoir column
        const float* xp = x + (m * NDIM + n) * TDIM + t;
        const float hv = tanhf(SLOPE * (v + *xp));
        hbuf[nxt * (BDIM * NDIM) + m * NDIM + n] = hv;
        if (t + 1 < TDIM) __builtin_prefetch(xp + 1, 0, 1);  // next x_t
      }
    } else if (t >= 1) {
      // project res_{t-1} (in hcur) into out[:, :, t-1]
      proj_tile(hcur, lw, lbv, out, c0, k0, wave, lane, lo, hi, t - 1, red);
    }

    grid_barrier(cnt, (unsigned)(t + 1) * (unsigned)NWG);
  }

  // last step's projection: res_{T-1} lives in hbuf[0] (T even)
  if (!isRec) {
    proj_tile(hbuf, lw, lbv, out, c0, k0, wave, lane, lo, hi, TDIM - 1, red);
  }
}

extern "C" void kernel_launch(void* const* d_in, const int* in_sizes, int n_in,
                              void* d_out, int out_size, void* d_ws,
                              size_t ws_size, hipStream_t stream) {
  const float* x  = (const float*)d_in[0];   // [16, 1024, 4096]
  const float* W  = (const float*)d_in[1];   // [1024, 1024]
  const float* lw = (const float*)d_in[2];   // [256, 1024]
  const float* lb = (const float*)d_in[3];   // [256]
  float* out = (float*)d_out;                // [16, 256, 4096]

  float* hbuf = (float*)d_ws;                // 2 * 16 * 1024 floats (128 KB)
  unsigned* cnt =
      (unsigned*)((char*)d_ws + (size_t)2 * BDIM * NDIM * sizeof(float));

  const int initN = 2 * BDIM * NDIM;
  rc_init<<<(initN + 255) / 256, 256, 0, stream>>>(hbuf, cnt);
  rc_fused<<<NWG, WGSIZE, 0, stream>>>(x, W, lw, lb, out, hbuf, cnt);
  (void)in_sizes; (void)n_in; (void)out_size; (void)ws_size;
}